// Wkv7_89730456748339
// MI455X (gfx1250) — compile-verified
//
#include <hip/hip_runtime.h>
#include <stdint.h>

// ---------------- problem constants ----------------
#define T_LEN 2048
#define H_N   64
#define D_DIM 64
#define L_CH  16                 // chunk length
#define N_CH  (T_LEN / L_CH)     // 128 chunks
#define R_SPL 2                  // state-row split per head
#define ROWS  (D_DIM / R_SPL)    // 32 state rows per block
#define NTHR  128                // 4 waves (wave32)

// ---------------- LDS layout (floats) ----------------
// stride 76: row-pair frag loads conflict-free (12*m mod 64 distinct),
// col-pair frag loads conflict-free (2*76 mod 64 = 24 > 15), 76*4 % 16 == 0
// so async b128 LDS writes stay 16B aligned.
#define SP 76
#define GP 17
#define SLAB_SZ   (L_CH * SP)                // one 16x64(+pad) input tile
#define SLABSET   (6 * SLAB_SZ)              // r,w,k,v,a,b for one chunk
#define OFF_SLAB  0                          // 2 slab sets (double buffer)
#define OFF_S     (2 * SLABSET)              // state rows: ROWS x SP
#define OFF_QL    (OFF_S + ROWS * SP)        // 64: end-of-chunk decay q_L
#define OFF_G     (OFF_QL + 64)              // 4 gram matrices 16x17
#define OFF_P     (OFF_G + 4 * L_CH * GP)    // P: 16 x SP
#define OFF_XO    (OFF_P + L_CH * SP)        // X partial: 16 x SP
#define LDS_FLOATS (OFF_XO + L_CH * SP)      // = 20608 floats = 82.4 KB

typedef float v2f __attribute__((ext_vector_type(2)));
typedef float v8f __attribute__((ext_vector_type(8)));

// ---------------- CDNA5 primitives ----------------
__device__ __forceinline__ v8f wmma4(v2f a, v2f b, v8f c) {
#if defined(__HIP_DEVICE_COMPILE__)
  // V_WMMA_F32_16X16X4_F32 : D = A(16x4) x B(4x16) + C(16x16), pure fp32
  return __builtin_amdgcn_wmma_f32_16x16x4_f32(false, a, false, b, (short)0, c,
                                               false, false);
#else
  (void)a; (void)b; return c;
#endif
}

__device__ __forceinline__ void async_b128(unsigned lds_off, const void* g) {
#if defined(__HIP_DEVICE_COMPILE__)
  unsigned long long ga = (unsigned long long)g;
  asm volatile("global_load_async_to_lds_b128 %0, %1, off"
               :: "v"(lds_off), "v"(ga) : "memory");
#else
  (void)lds_off; (void)g;
#endif
}

__device__ __forceinline__ void wait_async0() {
#if defined(__HIP_DEVICE_COMPILE__)
  asm volatile("s_wait_asynccnt 0" ::: "memory");
#endif
}

// ---------------- WMMA fragment gathers from LDS ----------------
// lane decomposition: m16 = lane&15, khi = lane>>4, k = kb + 2*khi
// A-frag from row-major M x K source:  A[m][k]   -> frag_row(p, ld, m,  k)
// A-frag from transposed source Q[k][m]          -> frag_col(p, ld, m,  k)
// B-frag from row-major K x N source:  B[k][n]   -> frag_col(p, ld, n,  k)
// B-frag from transposed source Q[n][k]          -> frag_row(p, ld, n,  k)
__device__ __forceinline__ v2f frag_row(const float* p, int ld, int row, int k) {
  v2f r; const float* q = p + row * ld + k;
  r.x = q[0]; r.y = q[1]; return r;
}
__device__ __forceinline__ v2f frag_col(const float* p, int ld, int col, int k) {
  v2f r; const float* q = p + k * ld + col;
  r.x = q[0]; r.y = q[ld]; return r;
}

// issue all async loads for one chunk into a slab set (16B-granular, aligned)
__device__ __forceinline__ void issue_chunk(const float* const* gin, float* slab,
                                            int c, int tid, int h) {
  for (int it = tid; it < 6 * L_CH * 4; it += NTHR) {
    int arr = it >> 6;        // which input array
    int rem = it & 63;
    int t   = rem >> 2;       // row in chunk
    int seg = rem & 3;        // 16B segment
    const float* g = gin[arr] + (size_t)(c * L_CH + t) * (H_N * D_DIM)
                              + h * D_DIM + seg * 16;
    unsigned lds_off = (unsigned)(unsigned long long)(const void*)
                       (slab + arr * SLAB_SZ + t * SP + seg * 16);
    async_b128(lds_off, (const void*)g);
  }
}

// one 16x16 state tile: S(i0..,j0..) = (S + P^T@Bbar + V^T@Kbar) * qL
__device__ __forceinline__ void state_tile(float* S, const float* P,
                                           const float* Bb, const float* Kb,
                                           const float* Vv, const float* QL,
                                           int i0, int j0, int ib,
                                           int m16, int khi) {
  v8f acc;
  #pragma unroll
  for (int g8 = 0; g8 < 8; ++g8)
    acc[g8] = S[(i0 + g8 + 8 * khi) * SP + j0 + m16];
  #pragma unroll
  for (int kb = 0; kb < L_CH; kb += 4) {
    int k = kb + 2 * khi;
    acc = wmma4(frag_col(P, SP, i0 + m16, k),            // A = P^T
                frag_col(Bb, SP, j0 + m16, k), acc);     // B = Bbar
    acc = wmma4(frag_col(Vv, SP, ib + i0 + m16, k),      // A = V^T
                frag_col(Kb, SP, j0 + m16, k), acc);     // B = Kbar
  }
  float ql = QL[j0 + m16];
  #pragma unroll
  for (int g8 = 0; g8 < 8; ++g8)
    S[(i0 + g8 + 8 * khi) * SP + j0 + m16] = acc[g8] * ql;
}

// ---------------- kernel ----------------
__global__ __launch_bounds__(NTHR)
void wkv7_chunk_kernel(const float* __restrict__ gr, const float* __restrict__ gw,
                       const float* __restrict__ gk, const float* __restrict__ gv,
                       const float* __restrict__ ga, const float* __restrict__ gb,
                       const float* __restrict__ gs0,
                       float* __restrict__ xout, float* __restrict__ sout)
{
  extern __shared__ float sm[];
  float* SLAB = sm + OFF_SLAB;   // 2 x [6][16][SP]  order: r,w,k,v,a,b
  float* S    = sm + OFF_S;      // [ROWS][SP]   state rows (local)
  float* QL   = sm + OFF_QL;     // [64]
  float* G    = sm + OFF_G;      // [4][16][GP]  0:B~ 1:K~ (strict) 2:RB 3:RK (tril)
  float* P    = sm + OFF_P;      // [16][SP]     (cols = local state row i)
  float* XO   = sm + OFF_XO;     // [16][SP]     partial outputs

  const int tid  = threadIdx.x;
  const int lane = tid & 31;
  const int wv   = tid >> 5;
  const int m16  = lane & 15;
  const int khi  = lane >> 4;

  const int h  = (int)blockIdx.x >> 1;          // head
  const int ib = ((int)blockIdx.x & 1) * ROWS;  // global state-row base

  const float* gin[6] = { gr, gw, gk, gv, ga, gb };

  // prefetch chunk 0 while we load the state
  issue_chunk(gin, SLAB, 0, tid, h);

  for (int idx = tid; idx < ROWS * D_DIM; idx += NTHR) {
    int i = idx >> 6, j = idx & 63;
    S[i * SP + j] = gs0[(size_t)h * (D_DIM * D_DIM) + (size_t)(ib + i) * D_DIM + j];
  }
  __syncthreads();

  for (int c = 0; c < N_CH; ++c) {
    float* CUR = SLAB + (c & 1) * SLABSET;
    float* NXT = SLAB + ((c + 1) & 1) * SLABSET;

    // wait for chunk c's async fill (only its ops are outstanding), sync all
    wait_async0();
    __syncthreads();

    // kick off chunk c+1 into the other buffer: overlaps all compute below
    if (c + 1 < N_CH) issue_chunk(gin, NXT, c + 1, tid, h);

    // ---- stage B: decay transform in place; thread j handles one column ----
    if (tid < D_DIM) {
      const int j = tid;
      float cum = 0.f, qp = 1.f;            // q_{t-1}, starts at 1
      float* Rm = CUR + 0 * SLAB_SZ;
      float* Wm = CUR + 1 * SLAB_SZ;
      float* Km = CUR + 2 * SLAB_SZ;
      float* Am = CUR + 4 * SLAB_SZ;
      float* Bm = CUR + 5 * SLAB_SZ;
      #pragma unroll
      for (int t = 0; t < L_CH; ++t) {
        cum += Wm[t * SP + j];
        float qc = __expf(cum);             // q_t
        float qi = __expf(-cum);            // 1/q_t
        Am[t * SP + j] *= qp;               // abar = a * q_{t-1}
        Bm[t * SP + j] *= qi;               // bbar = b / q_t
        Km[t * SP + j] *= qi;               // kbar = k / q_t
        Rm[t * SP + j] *= qc;               // rbar = r * q_t
        qp = qc;
      }
      QL[j] = qp;                           // q_L
    }
    __syncthreads();

    const float* Rb = CUR + 0 * SLAB_SZ;
    const float* Kb = CUR + 2 * SLAB_SZ;
    const float* Vv = CUR + 3 * SLAB_SZ;   // untouched v
    const float* Ab = CUR + 4 * SLAB_SZ;
    const float* Bb = CUR + 5 * SLAB_SZ;

    // ---- stage C: 4 gram matrices (16x16, K=64), one per wave ----
    {
      const float* As = (wv < 2) ? Ab : Rb;
      const float* Bs = ((wv & 1) == 0) ? Bb : Kb;
      v8f acc = {};
      #pragma unroll
      for (int kb = 0; kb < D_DIM; kb += 4) {
        int k = kb + 2 * khi;
        acc = wmma4(frag_row(As, SP, m16, k),   // A[t][j]
                    frag_row(Bs, SP, m16, k),   // B[j][u] = Bs[u][j] (transposed)
                    acc);
      }
      float* Gm = G + wv * (L_CH * GP);
      const bool strict = (wv < 2);
      #pragma unroll
      for (int g8 = 0; g8 < 8; ++g8) {
        int t = g8 + 8 * khi;                 // row of D tile
        int u = m16;                          // col of D tile
        float val = acc[g8];
        if (strict ? (u >= t) : (u > t)) val = 0.f;   // causal mask
        Gm[t * GP + u] = val;
      }
    }
    __syncthreads();

    // ---- stage D (waves 0,1): RHS = Abar@S0^T + K~@V  -> P
    // ---- stage E1 (waves 2,3): Rbar@S0^T -> XO (doesn't need P) ----
    if (wv < 2) {
      const int i0 = wv * 16;
      v8f acc = {};
      #pragma unroll
      for (int kb = 0; kb < D_DIM; kb += 4) {
        int k = kb + 2 * khi;
        acc = wmma4(frag_row(Ab, SP, m16, k),
                    frag_row(S, SP, i0 + m16, k),          // B = S0^T
                    acc);
      }
      const float* GKS = G + 1 * (L_CH * GP);
      #pragma unroll
      for (int kb = 0; kb < L_CH; kb += 4) {
        int k = kb + 2 * khi;
        acc = wmma4(frag_row(GKS, GP, m16, k),
                    frag_col(Vv, SP, ib + i0 + m16, k),    // V[u][global i]
                    acc);
      }
      #pragma unroll
      for (int g8 = 0; g8 < 8; ++g8)
        P[(g8 + 8 * khi) * SP + i0 + m16] = acc[g8];
    } else {
      const int i0 = (wv - 2) * 16;
      v8f acc = {};
      #pragma unroll
      for (int kb = 0; kb < D_DIM; kb += 4) {
        int k = kb + 2 * khi;
        acc = wmma4(frag_row(Rb, SP, m16, k),
                    frag_row(S, SP, i0 + m16, k),
                    acc);
      }
      #pragma unroll
      for (int g8 = 0; g8 < 8; ++g8)
        XO[(g8 + 8 * khi) * SP + i0 + m16] = acc[g8];
    }
    __syncthreads();

    // ---- forward substitution: P = (I - B~)^-1 RHS, register-resident ----
    if (tid < ROWS) {
      const float* GBS = G;     // gram 0, strictly lower
      const int i = tid;
      float pc[L_CH];
      #pragma unroll
      for (int t = 0; t < L_CH; ++t) pc[t] = P[t * SP + i];
      #pragma unroll
      for (int t = 1; t < L_CH; ++t) {
        float accp = pc[t];
        #pragma unroll
        for (int u = 0; u < t; ++u)
          accp += GBS[t * GP + u] * pc[u];      // broadcast SGPR-ish reads
        pc[t] = accp;
      }
      #pragma unroll
      for (int t = 1; t < L_CH; ++t) P[t * SP + i] = pc[t];
    }
    __syncthreads();

    // ---- stage E2 (waves 0,1): X += RB@P + RK@V, store; then 1 state tile
    // ---- stage F  (waves 2,3): remaining 3 state tiles per row band ----
    if (wv < 2) {
      const int i0 = wv * 16;
      const float* GRB = G + 2 * (L_CH * GP);
      const float* GRK = G + 3 * (L_CH * GP);
      v8f acc;
      #pragma unroll
      for (int g8 = 0; g8 < 8; ++g8)
        acc[g8] = XO[(g8 + 8 * khi) * SP + i0 + m16];
      #pragma unroll
      for (int kb = 0; kb < L_CH; kb += 4) {
        int k = kb + 2 * khi;
        acc = wmma4(frag_row(GRB, GP, m16, k),
                    frag_col(P, SP, i0 + m16, k), acc);
        acc = wmma4(frag_row(GRK, GP, m16, k),
                    frag_col(Vv, SP, ib + i0 + m16, k), acc);
      }
      #pragma unroll
      for (int g8 = 0; g8 < 8; ++g8) {
        int t = g8 + 8 * khi;
        xout[(size_t)(c * L_CH + t) * (H_N * D_DIM) + h * D_DIM + ib + i0 + m16]
            = acc[g8];
      }
      // balance: this wave also updates state tile (i0, j0=0)
      state_tile(S, P, Bb, Kb, Vv, QL, i0, 0, ib, m16, khi);
    } else {
      const int i0 = (wv - 2) * 16;          // this wave's S row band
      for (int j0 = 16; j0 < D_DIM; j0 += 16)
        state_tile(S, P, Bb, Kb, Vv, QL, i0, j0, ib, m16, khi);
    }
  } // chunks

  __syncthreads();
  for (int idx = tid; idx < ROWS * D_DIM; idx += NTHR) {
    int i = idx >> 6, j = idx & 63;
    sout[(size_t)h * (D_DIM * D_DIM) + (size_t)(ib + i) * D_DIM + j] = S[i * SP + j];
  }
}

// ---------------- host launcher ----------------
extern "C" void kernel_launch(void* const* d_in, const int* in_sizes, int n_in,
                              void* d_out, int out_size, void* d_ws, size_t ws_size,
                              hipStream_t stream) {
  (void)in_sizes; (void)n_in; (void)d_ws; (void)ws_size; (void)out_size;
  // dict order: seq_length, r, w, k, v, a, b, state2
  const float* r  = (const float*)d_in[1];
  const float* w  = (const float*)d_in[2];
  const float* k  = (const float*)d_in[3];
  const float* v  = (const float*)d_in[4];
  const float* a  = (const float*)d_in[5];
  const float* b  = (const float*)d_in[6];
  const float* s0 = (const float*)d_in[7];
  float* xout = (float*)d_out;                              // (T,H,1,D)
  float* sout = xout + (size_t)T_LEN * H_N * D_DIM;         // (H,D,D)

  dim3 grid(H_N * R_SPL);
  dim3 block(NTHR);
  size_t shmem = (size_t)LDS_FLOATS * sizeof(float);        // ~82 KB
  wkv7_chunk_kernel<<<grid, block, shmem, stream>>>(r, w, k, v, a, b, s0,
                                                    xout, sout);
}